// WeightLayer_31903017074762
// MI455X (gfx1250) — compile-verified
//
#include <hip/hip_runtime.h>
#include <stdint.h>

#define FULL   1280
#define NPIX   (FULL * FULL)   /* 1,638,400 */
#define IMG    256
#define BND    (FULL - IMG)    /* 1024 */
#define PPB    8               /* planes per block in the apply kernel */

typedef float v4f __attribute__((ext_vector_type(4)));

/* ---- CDNA5 async global->LDS path, feature-guarded -------------------- */
#if defined(__has_builtin)
# if __has_builtin(__builtin_amdgcn_global_load_async_to_lds_b128)
#  define HAVE_ASYNC_LD 1
# endif
# if __has_builtin(__builtin_amdgcn_s_wait_asynccnt)
#  define HAVE_WAIT_ASYNC 1
# endif
# if __has_builtin(__builtin_amdgcn_s_wait_dscnt)
#  define HAVE_WAIT_DS 1
# endif
#endif

#if defined(HAVE_ASYNC_LD)
/* Builtin signature (from clang diagnostic): arg0 is AS(1) ptr to
   'int __attribute__((vector_size(16)))', i.e. a b128 payload pointer. */
typedef int v4i_vs __attribute__((vector_size(16)));
typedef __attribute__((address_space(1))) v4i_vs g_v4i;
typedef __attribute__((address_space(3))) v4i_vs l_v4i;

/* cpol: gfx12-style CPol immediate, TH in [2:0]; 1 = TH_NT (non-temporal)
   so the 210 MB input stream does not evict the L2-resident C/B maps. */
__device__ __forceinline__ void async_ld_b128_nt(const void* gsrc, void* ldst) {
  __builtin_amdgcn_global_load_async_to_lds_b128(
      (g_v4i*)(uintptr_t)gsrc,
      (l_v4i*)(uint32_t)(uintptr_t)ldst,
      0, 1);
}
#endif

__device__ __forceinline__ void wait_async_le(int n) {
#if defined(HAVE_WAIT_ASYNC)
  if (n == 0) __builtin_amdgcn_s_wait_asynccnt(0);
  else        __builtin_amdgcn_s_wait_asynccnt(1);
#else
  if (n == 0) asm volatile("s_wait_asynccnt 0x0" ::: "memory");
  else        asm volatile("s_wait_asynccnt 0x1" ::: "memory");
#endif
  asm volatile("" ::: "memory");  /* compiler ordering fence for LDS reads */
}

__device__ __forceinline__ void wait_ds0() {
#if defined(HAVE_WAIT_DS)
  __builtin_amdgcn_s_wait_dscnt(0);
#else
  asm volatile("s_wait_dscnt 0x0" ::: "memory");
#endif
  asm volatile("" ::: "memory");
}

/* ---- Kernel 1: build fold maps C = fold(w), B = fold(b) --------------- */
/* Placements along an axis: {128k : 0<=k<n} plus the boundary strip 1024. */
__global__ __launch_bounds__(256)
void wl_build_maps(const float* __restrict__ w, const float* __restrict__ b,
                   const int* __restrict__ nh_p, const int* __restrict__ nw_p,
                   float* __restrict__ Cmap, float* __restrict__ Bmap) {
  int idx = blockIdx.x * 256 + threadIdx.x;
  if (idx >= NPIX) return;
  int y = idx / FULL;
  int x = idx - y * FULL;
  int n_h = nh_p[0];
  int n_w = nw_p[0];

  int ry[3], ncy = 0;
  {
    int q = y >> 7;
    for (int dk = 1; dk >= 0; --dk) {
      int k = q - dk;
      if (k >= 0 && k < n_h) ry[ncy++] = y - (k << 7);
    }
    if (y >= BND) ry[ncy++] = y - BND;  /* bottom boundary strip */
  }
  int rx[3], ncx = 0;
  {
    int q = x >> 7;
    for (int dk = 1; dk >= 0; --dk) {
      int k = q - dk;
      if (k >= 0 && k < n_w) rx[ncx++] = x - (k << 7);
    }
    if (x >= BND) rx[ncx++] = x - BND;  /* right boundary strip */
  }

  float cs = 0.f, bs = 0.f;
  for (int i = 0; i < ncy; ++i) {
    int row = ry[i] * IMG;
    for (int j = 0; j < ncx; ++j) {
      cs += w[row + rx[j]];
      bs += b[row + rx[j]];
    }
  }
  Cmap[idx] = cs;
  Bmap[idx] = bs;
}

/* ---- Kernel 2: out = (in*C + B) / C, streaming over planes ------------ */
/* One thread owns 4 consecutive pixels; C/B/1/C live in registers and are
   reused across PPB planes. Input stream is double-buffered through LDS via
   the CDNA5 async copy path (NT cache policy); output uses non-temporal
   b128 stores, so the 420 MB stream leaves the C/B maps resident in L2. */
__global__ __launch_bounds__(256)
void wl_apply(const float* __restrict__ in, const float* __restrict__ Cmap,
              const float* __restrict__ Bmap, float* __restrict__ out,
              int nplanes) {
  const int tid = threadIdx.x;
  const int g   = blockIdx.x * 256 + tid;          /* v4 group index */
  if (g >= NPIX / 4) return;
  const size_t e0 = (size_t)g * 4;                 /* element offset in plane */
  const int    p0 = blockIdx.y * PPB;

  v4f C  = *(const v4f*)(Cmap + e0);
  v4f Bv = *(const v4f*)(Bmap + e0);
  v4f R  = 1.0f / C;                               /* amortized exact divide */

#if defined(HAVE_ASYNC_LD)
  __shared__ v4f buf[2][256];
  if (p0 < nplanes)
    async_ld_b128_nt(in + (size_t)p0 * NPIX + e0, &buf[0][tid]);

  for (int n = 0; n < PPB; ++n) {
    const int plane = p0 + n;
    if (plane >= nplanes) break;
    const bool more = (n + 1 < PPB) && (plane + 1 < nplanes);
    if (more) {
      wait_ds0();  /* ensure last read of buf[(n+1)&1] has drained */
      async_ld_b128_nt(in + (size_t)(plane + 1) * NPIX + e0,
                       &buf[(n + 1) & 1][tid]);
    }
    wait_async_le(more ? 1 : 0);
    v4f v = buf[n & 1][tid];                       /* ds_load_b128 */
    v4f o = (v * C + Bv) * R;                      /* fma + mul per lane */
    __builtin_nontemporal_store(o, (v4f*)(out + (size_t)plane * NPIX + e0));
  }
#else
  for (int n = 0; n < PPB; ++n) {
    const int plane = p0 + n;
    if (plane >= nplanes) break;
    if (n + 1 < PPB && plane + 1 < nplanes)
      __builtin_prefetch(in + (size_t)(plane + 1) * NPIX + e0, 0, 1);
    v4f v = __builtin_nontemporal_load((const v4f*)(in + (size_t)plane * NPIX + e0));
    v4f o = (v * C + Bv) * R;
    __builtin_nontemporal_store(o, (v4f*)(out + (size_t)plane * NPIX + e0));
  }
#endif
}

/* ---- Launcher --------------------------------------------------------- */
extern "C" void kernel_launch(void* const* d_in, const int* in_sizes, int n_in,
                              void* d_out, int out_size, void* d_ws, size_t ws_size,
                              hipStream_t stream) {
  const float* in = (const float*)d_in[0];
  const float* w  = (const float*)d_in[1];
  const float* b  = (const float*)d_in[2];
  const int*   nh = (const int*)d_in[3];
  const int*   nw = (const int*)d_in[4];
  float* Cmap = (float*)d_ws;
  float* Bmap = Cmap + NPIX;
  float* out  = (float*)d_out;

  const int nplanes = out_size / NPIX;             /* 32 */

  wl_build_maps<<<dim3((NPIX + 255) / 256), dim3(256), 0, stream>>>(
      w, b, nh, nw, Cmap, Bmap);

  dim3 grid(NPIX / 4 / 256, (nplanes + PPB - 1) / PPB);  /* (1600, 4) */
  wl_apply<<<grid, dim3(256), 0, stream>>>(in, Cmap, Bmap, out, nplanes);
}